// BandSplit_49340584296744
// MI455X (gfx1250) — compile-verified
//
#include <hip/hip_runtime.h>

#define NBANDS 31
#define NB 8
#define NT 2000
#define NF 257
#define NCH 128
#define EPSV 1e-5f

typedef float v2f __attribute__((ext_vector_type(2)));
typedef float v8f __attribute__((ext_vector_type(8)));

struct Params {
    const float* x;        // (NB, 2, NF, NT)
    float*       out;      // (NB, NBANDS, NT, NCH)
    float*       bws;      // WMMA-fragment-layout normalized weights (K zero-padded)
    float*       biasws;   // fused bias, (NBANDS*NB, NCH)
    const float* nw[NBANDS];
    const float* nbias[NBANDS];
    const float* fcw[NBANDS];   // (NCH, d) row-major
    const float* fcb[NBANDS];
    int bofs[NBANDS];           // float offset of band's fragment block in bws
    int lo[NBANDS];
    int bw[NBANDS];
};

// Fragment layout (matches f32 WMMA B operand, 16x16x4):
//   offset(band,b,k,n) = bofs[band] + b*nsteps*512
//                      + ((step*8 + n>>4)*32 + (k&2 ? 16:0) + (n&15))*2 + (k&1)
// so a wave reads one contiguous v2f per lane per (step, ntile).

// ---- Kernel 1: per-(band,b) stats + fold GroupNorm into weights/bias + swizzle ----
__global__ __launch_bounds__(256) void bs_prep(Params p) {
    const int band = blockIdx.x / NB;
    const int b    = blockIdx.x % NB;
    const int bw = p.bw[band], lo = p.lo[band], d = bw * 2;
    const int nsteps = (d + 3) >> 2;
    const int tid = threadIdx.x;

    float s = 0.f, sq = 0.f;
    for (int k = 0; k < d; ++k) {
        const int f = k >> 1, c = k & 1;
        const float* row = p.x + ((size_t)(b * 2 + c) * NF + lo + f) * NT;
        for (int t = tid; t < NT; t += 256) {       // contiguous along t
            float v = row[t];
            s += v; sq += v * v;
        }
    }
    __shared__ float shs[256], shq[256];
    __shared__ float smu, srs;
    shs[tid] = s; shq[tid] = sq;
    __syncthreads();
    for (int off = 128; off > 0; off >>= 1) {
        if (tid < off) { shs[tid] += shs[tid + off]; shq[tid] += shq[tid + off]; }
        __syncthreads();
    }
    if (tid == 0) {
        const float n   = (float)(d * NT);
        const float mu  = shs[0] / n;
        const float var = shq[0] / n - mu * mu;     // biased variance (torch)
        smu = mu;
        srs = rsqrtf(fmaxf(var, 0.f) + EPSV);
    }
    __syncthreads();

    // Fold norm into weights; write B fragments (zero K-padding) + fused bias.
    if (tid < NCH) {
        const int n = tid;
        const float mu = smu, rs = srs;
        const float* fcw = p.fcw[band];
        float* Bf = p.bws + p.bofs[band] + (size_t)b * nsteps * 512;
        float acc = p.fcb[band][n];
        const int kpad = nsteps * 4;
        for (int k = 0; k < kpad; ++k) {
            float wp = 0.f;
            if (k < d) {
                const float sc = rs * p.nw[band][k];
                const float wv = fcw[(size_t)n * d + k];
                wp = wv * sc;
                acc += wv * (p.nbias[band][k] - mu * sc);
            }
            const int step = k >> 2;
            const int lane = ((k & 2) ? 16 : 0) + (n & 15);
            Bf[((size_t)(step * 8 + (n >> 4)) * 32 + lane) * 2 + (k & 1)] = wp;
        }
        p.biasws[(size_t)(band * NB + b) * NCH + n] = acc;
    }
}

// ---- Kernel 2: GEMM via V_WMMA_F32_16X16X4_F32 ----
// Block = 4 waves; covers (band, b, 16-t tile) x 128 N. Each wave: two 16x16 N-tiles.
__global__ __launch_bounds__(128) void bs_gemm(Params p) {
    const int lane = threadIdx.x & 31;
    const int wave = threadIdx.x >> 5;

    int g = blockIdx.x;
    const int ttile = g % (NT / 16); g /= (NT / 16);
    const int b     = g % NB;
    const int band  = g / NB;

    const int bw = p.bw[band], lo = p.lo[band], d = bw * 2;
    const int nsteps = (d + 3) >> 2;
    const int nfull  = d >> 2;
    const int t0 = ttile * 16;

    const int nlane = lane & 15;
    const int hi    = lane >> 4;            // 0: K{0,1}, 1: K{2,3} halves
    const int mrow  = t0 + nlane;

    // A pointers: a.x in c=0 plane, a.y in c=1 plane; f advances by 2 per step.
    const float* __restrict__ xb  = p.x + (size_t)b * 2 * NF * NT;
    const float* __restrict__ axp = xb + ((size_t)(lo + hi)) * NT + mrow;
    const float* __restrict__ ayp = axp + (size_t)NF * NT;

    // B fragments: contiguous v2f per lane per (step, ntile).
    const float* __restrict__ Bp =
        p.bws + p.bofs[band] + (size_t)b * nsteps * 512 + ((size_t)(wave * 2) * 32 + lane) * 2;

    v8f c0 = {}; v8f c1 = {};
    for (int s = 0; s < nfull; ++s) {
        v2f a;  a.x = *axp;  a.y = *ayp;
        v2f b0 = *(const v2f*)Bp;
        v2f b1 = *(const v2f*)(Bp + 64);
        c0 = __builtin_amdgcn_wmma_f32_16x16x4_f32(false, a, false, b0, (short)0, c0, false, false);
        c1 = __builtin_amdgcn_wmma_f32_16x16x4_f32(false, a, false, b1, (short)0, c1, false, false);
        axp += 2 * NT;  ayp += 2 * NT;  Bp += 512;
    }
    if (d & 2) {
        // Peel: low lanes need f = bw-1 (valid); high lanes clamp back one row,
        // their B columns are zero so the clamped value contributes nothing.
        const int back = hi ? NT : 0;
        v2f a;  a.x = *(axp - back);  a.y = *(ayp - back);
        v2f b0 = *(const v2f*)Bp;
        v2f b1 = *(const v2f*)(Bp + 64);
        c0 = __builtin_amdgcn_wmma_f32_16x16x4_f32(false, a, false, b0, (short)0, c0, false, false);
        c1 = __builtin_amdgcn_wmma_f32_16x16x4_f32(false, a, false, b1, (short)0, c1, false, false);
    }

    // Epilogue: fused bias + store. D layout: VGPR r -> M = r (+8 for hi lanes), N = lane&15.
    const float* biasp = p.biasws + (size_t)(band * NB + b) * NCH;
    float* outp = p.out + ((size_t)b * NBANDS + band) * NT * NCH;
    const int n0 = wave * 32;
    const float bias0 = biasp[n0 + nlane];
    const float bias1 = biasp[n0 + 16 + nlane];
    const int rowoff = hi * 8;
#pragma unroll
    for (int r = 0; r < 8; ++r) {
        const int m = t0 + rowoff + r;
        outp[(size_t)m * NCH + n0 + nlane]      = c0[r] + bias0;
        outp[(size_t)m * NCH + n0 + 16 + nlane] = c1[r] + bias1;
    }
}

extern "C" void kernel_launch(void* const* d_in, const int* in_sizes, int n_in,
                              void* d_out, int out_size, void* d_ws, size_t ws_size,
                              hipStream_t stream) {
    static const int BW[NBANDS] = {3,3,3,3,3,3,3,3,3,3,
                                   6,6,6,6,6,6,6,6,6,6,6,6,
                                   16,16,16,16,16,16,16,16,
                                   27};
    Params p;
    p.x   = (const float*)d_in[0];
    p.out = (float*)d_out;
    int lo = 0, ofs = 0;
    for (int i = 0; i < NBANDS; ++i) {
        p.nw[i]    = (const float*)d_in[1 + i];
        p.nbias[i] = (const float*)d_in[1 + NBANDS + i];
        p.fcw[i]   = (const float*)d_in[1 + 2 * NBANDS + i];
        p.fcb[i]   = (const float*)d_in[1 + 3 * NBANDS + i];
        p.lo[i] = lo;
        p.bw[i] = BW[i];
        lo += BW[i];
        p.bofs[i] = ofs;
        const int d = BW[i] * 2;
        const int nsteps = (d + 3) >> 2;
        ofs += NB * nsteps * 512;               // fragment floats for this band (all samples)
    }
    p.bws    = (float*)d_ws;                    // ~2.2 MB of fragments
    p.biasws = p.bws + ofs;                     // + 248*128 floats of fused bias

    bs_prep<<<NBANDS * NB, 256, 0, stream>>>(p);
    bs_gemm<<<NBANDS * NB * (NT / 16), 128, 0, stream>>>(p);
}